// GAT_53541062312413
// MI455X (gfx1250) — compile-verified
//
#include <hip/hip_runtime.h>

typedef __attribute__((ext_vector_type(2))) float v2f;
typedef __attribute__((ext_vector_type(8))) float v8f;

#define NEG_SLOPE 0.2f

__device__ __forceinline__ unsigned f2key(float x) {
  unsigned u = __float_as_uint(x);
  return (u & 0x80000000u) ? ~u : (u | 0x80000000u);
}
__device__ __forceinline__ float key2f(unsigned k) {
  return (k & 0x80000000u) ? __uint_as_float(k & 0x7fffffffu) : __uint_as_float(~k);
}
__device__ __forceinline__ float lrelu(float v) { return v > 0.f ? v : NEG_SLOPE * v; }

template<int HC>
__device__ __forceinline__ void loadrow(const float* __restrict__ p, float* v) {
  if (HC % 4 == 0) {
    const float4* p4 = (const float4*)p;
#pragma unroll
    for (int i = 0; i < HC / 4; ++i) {
      float4 t = p4[i];
      v[4 * i + 0] = t.x; v[4 * i + 1] = t.y; v[4 * i + 2] = t.z; v[4 * i + 3] = t.w;
    }
  } else {
#pragma unroll
    for (int i = 0; i < HC; ++i) v[i] = p[i];
  }
}

// ---------------------------------------------------------------------------
// Dense node transform Y[N][OUT] = X[N][K] @ W[K][OUT] + B, via f32 WMMA.
// One wave = 16 rows. K in {4,16}; OUT <= 16 (zero-padded B columns).
// A 16x4 f32 layout: lanes0-15 {K0,K1}, lanes16-31 {K2,K3}.
// B 4x16 f32 layout: lanes0-15 K in {0,1} across V0,V1; lanes16-31 K in {2,3}.
// C/D: VGPR j -> M=j (lanes0-15), M=j+8 (lanes16-31); N = lane&15.
// ---------------------------------------------------------------------------
template<int K, int OUT>
__global__ void wmma_transform(const float* __restrict__ X, const float* __restrict__ W,
                               const float* __restrict__ B, float* __restrict__ Y, int N) {
  int wave = (int)((blockIdx.x * blockDim.x + threadIdx.x) >> 5);
  int lane = (int)(threadIdx.x & 31);
  int base = wave * 16;
  if (base >= N) return;               // wave-uniform: EXEC stays all-ones for WMMA
  int m  = lane & 15;                  // A-row within tile / B,C column
  int kh = lane >> 4;                  // K-half select
  v8f c = {};
#pragma unroll
  for (int kk = 0; kk < K; kk += 4) {
    int k0 = kk + 2 * kh;
    v2f a, b;
    const float* xp = X + (size_t)(base + m) * K + k0;
    a.x = xp[0];
    a.y = xp[1];
    b.x = (m < OUT) ? W[(size_t)(k0 + 0) * OUT + m] : 0.f;
    b.y = (m < OUT) ? W[(size_t)(k0 + 1) * OUT + m] : 0.f;
    c = __builtin_amdgcn_wmma_f32_16x16x4_f32(false, a, false, b, (short)0, c, false, false);
  }
  float bias = (m < OUT) ? B[m] : 0.f;
#pragma unroll
  for (int j = 0; j < 8; ++j) {
    int row = base + j + 8 * kh;
    if (m < OUT) Y[(size_t)row * OUT + m] = c[j] + bias;
  }
}

// ---------------------------------------------------------------------------
// Per-dst edge-weight stats (for self-loop fill_value='mean')
// ---------------------------------------------------------------------------
__global__ void deg_kernel(const int* __restrict__ dst, const float* __restrict__ w,
                           float* __restrict__ cnt, float* __restrict__ wsum, int E) {
  int e = (int)(blockIdx.x * blockDim.x + threadIdx.x);
  if (e >= E) return;
  int d = dst[e];
  atomicAdd(&cnt[d], 1.f);
  atomicAdd(&wsum[d], w[e]);
}

__global__ void mean_kernel(float* __restrict__ wmean, const float* __restrict__ cnt, int N) {
  int n = (int)(blockIdx.x * blockDim.x + threadIdx.x);
  if (n >= N) return;
  wmean[n] = wmean[n] / fmaxf(cnt[n], 1.f);
}

// ---------------------------------------------------------------------------
// Self-loop prep: initialize mkey with the self-loop logit key (every node has
// exactly one self loop -> serves as segment-max init), zero denom/accum.
// ---------------------------------------------------------------------------
template<int H, int C>
__global__ void self_prep(const float* __restrict__ xl, const float* __restrict__ xr,
                          const float* __restrict__ wmean,
                          const float* __restrict__ we, const float* __restrict__ att,
                          unsigned* __restrict__ mkey, float* __restrict__ denom,
                          float* __restrict__ accum, int N) {
  int n = (int)(blockIdx.x * blockDim.x + threadIdx.x);
  if (n >= N) return;
  const int HC = H * C;
  float xlv[HC], xrv[HC];
  loadrow<HC>(xl + (size_t)n * HC, xlv);
  loadrow<HC>(xr + (size_t)n * HC, xrv);
  float wm = wmean[n];
#pragma unroll
  for (int h = 0; h < H; ++h) {
    float lg = 0.f;
#pragma unroll
    for (int c2 = 0; c2 < C; ++c2) {
      int i = h * C + c2;
      lg += lrelu(xlv[i] + xrv[i] + wm * we[i]) * att[i];
    }
    mkey[(size_t)n * H + h] = f2key(lg);
    denom[(size_t)n * H + h] = 0.f;
  }
#pragma unroll
  for (int i = 0; i < HC; ++i) accum[(size_t)n * HC + i] = 0.f;
}

// ---------------------------------------------------------------------------
// Edge pass 1: logits -> segment max (monotonic uint key atomicMax)
// ---------------------------------------------------------------------------
template<int H, int C>
__global__ void edge_max(const int* __restrict__ src, const int* __restrict__ dst,
                         const float* __restrict__ w,
                         const float* __restrict__ xl, const float* __restrict__ xr,
                         const float* __restrict__ we, const float* __restrict__ att,
                         unsigned* __restrict__ mkey, int E) {
  int e = (int)(blockIdx.x * blockDim.x + threadIdx.x);
  if (e >= E) return;
  const int HC = H * C;
  int s = src[e], d = dst[e];
  float wv = w[e];
  float xlv[HC], xrv[HC];
  loadrow<HC>(xl + (size_t)s * HC, xlv);
  loadrow<HC>(xr + (size_t)d * HC, xrv);
#pragma unroll
  for (int h = 0; h < H; ++h) {
    float lg = 0.f;
#pragma unroll
    for (int c2 = 0; c2 < C; ++c2) {
      int i = h * C + c2;
      lg += lrelu(xlv[i] + xrv[i] + wv * we[i]) * att[i];
    }
    atomicMax(&mkey[(size_t)d * H + h], f2key(lg));
  }
}

// ---------------------------------------------------------------------------
// Edge pass 2: p = exp(logit - max); accumulate unnormalized softmax sum and
// weighted messages (normalized in finalize -> single fused edge pass)
// ---------------------------------------------------------------------------
template<int H, int C>
__global__ void edge_accum(const int* __restrict__ src, const int* __restrict__ dst,
                           const float* __restrict__ w,
                           const float* __restrict__ xl, const float* __restrict__ xr,
                           const float* __restrict__ we, const float* __restrict__ att,
                           const unsigned* __restrict__ mkey,
                           float* __restrict__ denom, float* __restrict__ accum, int E) {
  int e = (int)(blockIdx.x * blockDim.x + threadIdx.x);
  if (e >= E) return;
  const int HC = H * C;
  int s = src[e], d = dst[e];
  float wv = w[e];
  float xlv[HC], xrv[HC];
  loadrow<HC>(xl + (size_t)s * HC, xlv);
  loadrow<HC>(xr + (size_t)d * HC, xrv);
#pragma unroll
  for (int h = 0; h < H; ++h) {
    float lg = 0.f;
#pragma unroll
    for (int c2 = 0; c2 < C; ++c2) {
      int i = h * C + c2;
      lg += lrelu(xlv[i] + xrv[i] + wv * we[i]) * att[i];
    }
    float m = key2f(mkey[(size_t)d * H + h]);
    float p = __expf(lg - m);
    atomicAdd(&denom[(size_t)d * H + h], p);
#pragma unroll
    for (int c2 = 0; c2 < C; ++c2) {
      int i = h * C + c2;
      atomicAdd(&accum[(size_t)d * HC + i], p * xlv[i]);
    }
  }
}

// ---------------------------------------------------------------------------
// Finalize: add self-loop term, normalize, add bias, combine into out:
//   RELU=true : out[n][i] += relu(gat)   (layer residual path)
//   RELU=false: out[n][i] += gat         (final scalar layer)
// ---------------------------------------------------------------------------
template<int H, int C, bool RELU>
__global__ void finalize(const float* __restrict__ xl, const float* __restrict__ xr,
                         const float* __restrict__ wmean,
                         const unsigned* __restrict__ mkey, const float* __restrict__ denom,
                         const float* __restrict__ accum,
                         const float* __restrict__ we, const float* __restrict__ att,
                         const float* __restrict__ bias, float* __restrict__ out, int N) {
  int n = (int)(blockIdx.x * blockDim.x + threadIdx.x);
  if (n >= N) return;
  const int HC = H * C;
  float xlv[HC], xrv[HC];
  loadrow<HC>(xl + (size_t)n * HC, xlv);
  loadrow<HC>(xr + (size_t)n * HC, xrv);
  float wm = wmean[n];
#pragma unroll
  for (int h = 0; h < H; ++h) {
    float lg = 0.f;
#pragma unroll
    for (int c2 = 0; c2 < C; ++c2) {
      int i = h * C + c2;
      lg += lrelu(xlv[i] + xrv[i] + wm * we[i]) * att[i];
    }
    float m = key2f(mkey[(size_t)n * H + h]);
    float p = __expf(lg - m);
    float D = denom[(size_t)n * H + h] + p;
#pragma unroll
    for (int c2 = 0; c2 < C; ++c2) {
      int i = h * C + c2;
      float A = accum[(size_t)n * HC + i] + p * xlv[i];
      float g = A / D + bias[i];
      if (RELU) g = fmaxf(g, 0.f);
      out[(size_t)n * HC + i] += g;
    }
  }
}

// ---------------------------------------------------------------------------
extern "C" void kernel_launch(void* const* d_in, const int* in_sizes, int n_in,
                              void* d_out, int out_size, void* d_ws, size_t ws_size,
                              hipStream_t stream) {
  (void)n_in; (void)out_size; (void)ws_size;
  const float* x      = (const float*)d_in[0];
  const int*   e_src  = (const int*)d_in[1];
  const int*   e_dst  = (const int*)d_in[2];
  const float* w      = (const float*)d_in[3];
  const float* lin1_w = (const float*)d_in[4];  const float* lin1_b = (const float*)d_in[5];
  const float* lin2_w = (const float*)d_in[6];  const float* lin2_b = (const float*)d_in[7];
  const float* g1_wl  = (const float*)d_in[8];  const float* g1_bl  = (const float*)d_in[9];
  const float* g1_wr  = (const float*)d_in[10]; const float* g1_br  = (const float*)d_in[11];
  const float* g1_we  = (const float*)d_in[12]; const float* g1_att = (const float*)d_in[13];
  const float* g1_bias= (const float*)d_in[14];
  const float* gm_wl  = (const float*)d_in[15]; const float* gm_bl  = (const float*)d_in[16];
  const float* gm_wr  = (const float*)d_in[17]; const float* gm_br  = (const float*)d_in[18];
  const float* gm_we  = (const float*)d_in[19]; const float* gm_att = (const float*)d_in[20];
  const float* gm_bias= (const float*)d_in[21];
  const float* g2_wl  = (const float*)d_in[22]; const float* g2_bl  = (const float*)d_in[23];
  const float* g2_wr  = (const float*)d_in[24]; const float* g2_br  = (const float*)d_in[25];
  const float* g2_we  = (const float*)d_in[26]; const float* g2_att = (const float*)d_in[27];
  const float* g2_bias= (const float*)d_in[28];

  const int N = in_sizes[0] / 4;
  const int E = in_sizes[1];

  // workspace layout (floats); ~74*N floats total (~30 MB) -> L2-resident
  float* ws = (float*)d_ws;
  size_t o = 0;
  float*    cnt   = ws + o; o += (size_t)N;
  float*    wmean = ws + o; o += (size_t)N;        // wsum then mean in-place
  float*    h     = ws + o; o += (size_t)N * 16;   // hidden state / lin1 base
  float*    xl    = ws + o; o += (size_t)N * 16;
  float*    xr    = ws + o; o += (size_t)N * 16;
  unsigned* mkey  = (unsigned*)(ws + o); o += (size_t)N * 4;
  float*    denom = ws + o; o += (size_t)N * 4;
  float*    accum = ws + o; o += (size_t)N * 16;
  float*    outf  = (float*)d_out;

  const int TB = 256;
  dim3 blk(TB);
  dim3 gE((E + TB - 1) / TB);
  dim3 gN((N + TB - 1) / TB);
  int waves = (N + 15) / 16;                       // one wave per 16-row tile
  dim3 gT(((size_t)waves * 32 + TB - 1) / TB);

  // --- per-dst edge-weight mean (shared by all layers) ---
  hipMemsetAsync(d_ws, 0, (size_t)2 * N * sizeof(float), stream);
  deg_kernel<<<gE, blk, 0, stream>>>(e_dst, w, cnt, wmean, E);
  mean_kernel<<<gN, blk, 0, stream>>>(wmean, cnt, N);

  // --- layer 1: h = x@lin1 + relu(gat1(x))  (4 -> 16, H=4, C=4) ---
  wmma_transform<4, 16><<<gT, blk, 0, stream>>>(x, lin1_w, lin1_b, h, N);
  wmma_transform<4, 16><<<gT, blk, 0, stream>>>(x, g1_wl, g1_bl, xl, N);
  wmma_transform<4, 16><<<gT, blk, 0, stream>>>(x, g1_wr, g1_br, xr, N);
  self_prep<4, 4><<<gN, blk, 0, stream>>>(xl, xr, wmean, g1_we, g1_att, mkey, denom, accum, N);
  edge_max<4, 4><<<gE, blk, 0, stream>>>(e_src, e_dst, w, xl, xr, g1_we, g1_att, mkey, E);
  edge_accum<4, 4><<<gE, blk, 0, stream>>>(e_src, e_dst, w, xl, xr, g1_we, g1_att, mkey, denom, accum, E);
  finalize<4, 4, true><<<gN, blk, 0, stream>>>(xl, xr, wmean, mkey, denom, accum, g1_we, g1_att, g1_bias, h, N);

  // --- mid layer: h = h + relu(gatm(h))  (16 -> 16, H=4, C=4) ---
  wmma_transform<16, 16><<<gT, blk, 0, stream>>>(h, gm_wl, gm_bl, xl, N);
  wmma_transform<16, 16><<<gT, blk, 0, stream>>>(h, gm_wr, gm_br, xr, N);
  self_prep<4, 4><<<gN, blk, 0, stream>>>(xl, xr, wmean, gm_we, gm_att, mkey, denom, accum, N);
  edge_max<4, 4><<<gE, blk, 0, stream>>>(e_src, e_dst, w, xl, xr, gm_we, gm_att, mkey, E);
  edge_accum<4, 4><<<gE, blk, 0, stream>>>(e_src, e_dst, w, xl, xr, gm_we, gm_att, mkey, denom, accum, E);
  finalize<4, 4, true><<<gN, blk, 0, stream>>>(xl, xr, wmean, mkey, denom, accum, gm_we, gm_att, gm_bias, h, N);

  // --- final: out = h@lin2 + gat2(h)  (16 -> 1, H=1, C=1, mean==identity) ---
  wmma_transform<16, 1><<<gT, blk, 0, stream>>>(h, lin2_w, lin2_b, outf, N);
  wmma_transform<16, 1><<<gT, blk, 0, stream>>>(h, g2_wl, g2_bl, xl, N);
  wmma_transform<16, 1><<<gT, blk, 0, stream>>>(h, g2_wr, g2_br, xr, N);
  self_prep<1, 1><<<gN, blk, 0, stream>>>(xl, xr, wmean, g2_we, g2_att, mkey, denom, accum, N);
  edge_max<1, 1><<<gE, blk, 0, stream>>>(e_src, e_dst, w, xl, xr, g2_we, g2_att, mkey, E);
  edge_accum<1, 1><<<gE, blk, 0, stream>>>(e_src, e_dst, w, xl, xr, g2_we, g2_att, mkey, denom, accum, E);
  finalize<1, 1, false><<<gN, blk, 0, stream>>>(xl, xr, wmean, mkey, denom, accum, g2_we, g2_att, g2_bias, outf, N);
}